// VN_Ori_Globa6D_22067541967082
// MI455X (gfx1250) — compile-verified
//
#include <hip/hip_runtime.h>
#include <math.h>

// ---------------- constants (fixed by the reference) ----------------
#define BB     4
#define NN     2048
#define KNEI   20
#define FF     1024
#define EPSV   1e-6f
#define BN_EPSV 1e-5f
#define NEGS   0.2f

typedef __attribute__((ext_vector_type(2))) float v2f;
typedef __attribute__((ext_vector_type(8))) float v8f;

__device__ inline v8f wmma_f32_k4(v2f a, v2f b, v8f c) {
#if defined(__HIP_DEVICE_COMPILE__)
  // (neg_a, A, neg_b, B, c_mod, C, reuse_a, reuse_b)
  return __builtin_amdgcn_wmma_f32_16x16x4_f32(false, a, false, b, (short)0, c, false, false);
#else
  (void)a; (void)b;
  return c;
#endif
}

// ---------------- squared norms per point ----------------
__global__ void sqnorm_kernel(const float* __restrict__ x2d, float* __restrict__ xx,
                              int D, int N, int total) {
  int gid = blockIdx.x * blockDim.x + threadIdx.x;
  if (gid >= total) return;
  int n = gid % N, b = gid / N;
  const float* xb = x2d + (size_t)b * D * N;
  float s = 0.f;
  for (int d = 0; d < D; ++d) { float v = xb[(size_t)d * N + n]; s += v * v; }
  xx[gid] = s;
}

// ---------------- KNN via WMMA f32 16x16x4 Gram tiles + per-row top-20 ----------------
__global__ void knn_topk_kernel(const float* __restrict__ x2d, const float* __restrict__ xx,
                                int* __restrict__ idxout, int D, int N) {
  __shared__ float ldsA[16][64];
  __shared__ float ldsB[16][64];
  __shared__ float ldsC[16][16];
  __shared__ float colxx[16];
  int rowTiles = N >> 4;
  int b  = blockIdx.x / rowTiles;
  int m0 = (blockIdx.x % rowTiles) << 4;
  int lane = threadIdx.x;
  const float* xb = x2d + (size_t)b * D * N;

  for (int t = lane; t < 16 * 64; t += 32) {
    int k = t >> 4, m = t & 15;
    ldsA[m][k] = (k < D) ? xb[(size_t)k * N + m0 + m] : 0.f;
  }
  float rowxx = (lane < 16) ? xx[(size_t)b * N + m0 + lane] : 0.f;

  float best[KNEI]; int bidx[KNEI];
#pragma unroll
  for (int j = 0; j < KNEI; ++j) { best[j] = -3.0e38f; bidx[j] = 0; }

  int ksteps = (D + 3) >> 2;
  int row  = lane & 15;
  int koff = (lane >> 4) << 1;
  __syncthreads();

  for (int c0 = 0; c0 < N; c0 += 16) {
    for (int t = lane; t < 16 * 64; t += 32) {
      int k = t >> 4, c = t & 15;
      ldsB[c][k] = (k < D) ? xb[(size_t)k * N + c0 + c] : 0.f;
    }
    if (lane < 16) colxx[lane] = xx[(size_t)b * N + c0 + lane];
    __syncthreads();

    v8f acc = {};
    for (int kk = 0; kk < ksteps; ++kk) {
      v2f a, bb;
      a.x  = ldsA[row][4 * kk + koff];  a.y  = ldsA[row][4 * kk + koff + 1];
      bb.x = ldsB[row][4 * kk + koff];  bb.y = ldsB[row][4 * kk + koff + 1];
      acc = wmma_f32_k4(a, bb, acc);
    }
#pragma unroll
    for (int r = 0; r < 8; ++r) {
      int rr = r + ((lane >= 16) ? 8 : 0);
      ldsC[rr][lane & 15] = acc[r];
    }
    __syncthreads();

    if (lane < 16) {
#pragma unroll
      for (int c = 0; c < 16; ++c) {
        float nd = 2.f * ldsC[lane][c] - rowxx - colxx[c];
        if (nd > best[KNEI - 1]) {
          best[KNEI - 1] = nd; bidx[KNEI - 1] = c0 + c;
#pragma unroll
          for (int j = KNEI - 1; j > 0; --j) {
            if (best[j] > best[j - 1]) {
              float tv = best[j]; best[j] = best[j - 1]; best[j - 1] = tv;
              int   ti = bidx[j]; bidx[j] = bidx[j - 1]; bidx[j - 1] = ti;
            }
          }
        }
      }
    }
    __syncthreads();
  }
  if (lane < 16) {
    size_t base = ((size_t)b * N + m0 + lane) * KNEI;
#pragma unroll
    for (int j = 0; j < KNEI; ++j) idxout[base + j] = bidx[j];
  }
}

// ---------------- VN stage 1-5: BN stats (deterministic 2-level reduction) ----------------
// graph-feature input: src (B,C,3,N), feat = [nbr-ctr ; ctr]
template <int O, int C>
__global__ void vn_stats_gf(const float* __restrict__ src, const int* __restrict__ idx,
                            const float* __restrict__ wf, float* __restrict__ part,
                            int N, int P, int chunkSize) {
  const int C2 = 2 * C;
  __shared__ float swf[C2];
  __shared__ float r1[256], r2[256];
  int o = blockIdx.y;
  for (int t = threadIdx.x; t < C2; t += blockDim.x) swf[t] = wf[o * C2 + t];
  __syncthreads();
  float s1 = 0.f, s2 = 0.f;
  int p0 = blockIdx.x * chunkSize;
  for (int i = threadIdx.x; i < chunkSize; i += blockDim.x) {
    int pos = p0 + i;
    if (pos < P) {
      int n = (pos / KNEI) % N;
      int b = pos / (KNEI * N);
      int nbr = idx[pos];
      float px = 0, py = 0, pz = 0;
#pragma unroll
      for (int c = 0; c < C; ++c) {
        const float* sc = src + ((size_t)(b * C + c) * 3) * N;
        float c0 = sc[n],       c1 = sc[N + n],   c2 = sc[2 * N + n];
        float n0 = sc[nbr],     n1 = sc[N + nbr], n2 = sc[2 * N + nbr];
        float wa = swf[c], wb = swf[C + c];
        px += wa * (n0 - c0) + wb * c0;
        py += wa * (n1 - c1) + wb * c1;
        pz += wa * (n2 - c2) + wb * c2;
      }
      float nrm = sqrtf(px * px + py * py + pz * pz) + EPSV;
      s1 += nrm; s2 += nrm * nrm;
    }
  }
  r1[threadIdx.x] = s1; r2[threadIdx.x] = s2; __syncthreads();
  for (int s = 128; s > 0; s >>= 1) {
    if (threadIdx.x < s) { r1[threadIdx.x] += r1[threadIdx.x + s]; r2[threadIdx.x] += r2[threadIdx.x + s]; }
    __syncthreads();
  }
  if (threadIdx.x == 0) {
    part[((size_t)o * gridDim.x + blockIdx.x) * 2 + 0] = r1[0];
    part[((size_t)o * gridDim.x + blockIdx.x) * 2 + 1] = r2[0];
  }
}

// direct input: in (B,C,3,N,KNEI)
template <int O, int C>
__global__ void vn_stats_direct(const float* __restrict__ in, const float* __restrict__ wf,
                                float* __restrict__ part, int N, int P, int chunkSize) {
  __shared__ float swf[C];
  __shared__ float r1[256], r2[256];
  int o = blockIdx.y;
  for (int t = threadIdx.x; t < C; t += blockDim.x) swf[t] = wf[o * C + t];
  __syncthreads();
  float s1 = 0.f, s2 = 0.f;
  int p0 = blockIdx.x * chunkSize;
  const size_t dstr = (size_t)N * KNEI;
  for (int i = threadIdx.x; i < chunkSize; i += blockDim.x) {
    int pos = p0 + i;
    if (pos < P) {
      int kk = pos % KNEI;
      int n  = (pos / KNEI) % N;
      int b  = pos / (KNEI * N);
      float px = 0, py = 0, pz = 0;
#pragma unroll
      for (int c = 0; c < C; ++c) {
        size_t base = (((size_t)(b * C + c) * 3) * N + n) * KNEI + kk;
        float w = swf[c];
        px += w * in[base];
        py += w * in[base + dstr];
        pz += w * in[base + 2 * dstr];
      }
      float nrm = sqrtf(px * px + py * py + pz * pz) + EPSV;
      s1 += nrm; s2 += nrm * nrm;
    }
  }
  r1[threadIdx.x] = s1; r2[threadIdx.x] = s2; __syncthreads();
  for (int s = 128; s > 0; s >>= 1) {
    if (threadIdx.x < s) { r1[threadIdx.x] += r1[threadIdx.x + s]; r2[threadIdx.x] += r2[threadIdx.x + s]; }
    __syncthreads();
  }
  if (threadIdx.x == 0) {
    part[((size_t)o * gridDim.x + blockIdx.x) * 2 + 0] = r1[0];
    part[((size_t)o * gridDim.x + blockIdx.x) * 2 + 1] = r2[0];
  }
}

__global__ void finalize_stats(const float* __restrict__ part, float* __restrict__ stats,
                               int O, int nchunks, float cntInv) {
  int o = blockIdx.x * blockDim.x + threadIdx.x;
  if (o >= O) return;
  float s1 = 0.f, s2 = 0.f;
  for (int c = 0; c < nchunks; ++c) {
    s1 += part[((size_t)o * nchunks + c) * 2 + 0];
    s2 += part[((size_t)o * nchunks + c) * 2 + 1];
  }
  float mu = s1 * cntInv;
  float var = s2 * cntInv - mu * mu;
  stats[o] = mu;
  stats[O + o] = rsqrtf(var + BN_EPSV);
}

// ---------------- VN leaky-relu apply kernels ----------------
__device__ inline void vn_lrelu3(float px, float py, float pz,   // batchnormed p
                                 float dx, float dy, float dz,
                                 float& ox, float& oy, float& oz) {
  float dot = px * dx + py * dy + pz * dz;
  float d2  = dx * dx + dy * dy + dz * dz;
  float qx, qy, qz;
  if (dot >= 0.f) { qx = px; qy = py; qz = pz; }
  else { float t = dot / (d2 + EPSV); qx = px - t * dx; qy = py - t * dy; qz = pz - t * dz; }
  ox = NEGS * px + (1.f - NEGS) * qx;
  oy = NEGS * py + (1.f - NEGS) * qy;
  oz = NEGS * pz + (1.f - NEGS) * qz;
}

// graph-feature input, write full (B,O,3,N,KNEI)
template <int O, int C>
__global__ void vn_apply_gf_full(const float* __restrict__ src, const int* __restrict__ idx,
                                 const float* __restrict__ wf, const float* __restrict__ wd,
                                 const float* __restrict__ stats, float* __restrict__ out,
                                 int N, int P) {
  const int C2 = 2 * C;
  __shared__ float swf[C2], swd[C2];
  __shared__ float smu, sistd;
  int o = blockIdx.y;
  for (int t = threadIdx.x; t < C2; t += blockDim.x) { swf[t] = wf[o * C2 + t]; swd[t] = wd[o * C2 + t]; }
  if (threadIdx.x == 0) { smu = stats[o]; sistd = stats[O + o]; }
  __syncthreads();
  int pos = blockIdx.x * blockDim.x + threadIdx.x;
  if (pos >= P) return;
  int kk = pos % KNEI;
  int n  = (pos / KNEI) % N;
  int b  = pos / (KNEI * N);
  int nbr = idx[pos];
  float px = 0, py = 0, pz = 0, dx = 0, dy = 0, dz = 0;
#pragma unroll
  for (int c = 0; c < C; ++c) {
    const float* sc = src + ((size_t)(b * C + c) * 3) * N;
    float c0 = sc[n],   c1 = sc[N + n],   c2 = sc[2 * N + n];
    float f0 = sc[nbr] - c0, f1 = sc[N + nbr] - c1, f2 = sc[2 * N + nbr] - c2;
    float wa = swf[c], wb = swf[C + c], va = swd[c], vb = swd[C + c];
    px += wa * f0 + wb * c0; py += wa * f1 + wb * c1; pz += wa * f2 + wb * c2;
    dx += va * f0 + vb * c0; dy += va * f1 + vb * c1; dz += va * f2 + vb * c2;
  }
  float nrm = sqrtf(px * px + py * py + pz * pz) + EPSV;
  float sc2 = (nrm - smu) * sistd / nrm;
  px *= sc2; py *= sc2; pz *= sc2;
  float ox, oy, oz;
  vn_lrelu3(px, py, pz, dx, dy, dz, ox, oy, oz);
  const size_t dstr = (size_t)N * KNEI;
  size_t base = (((size_t)(b * O + o) * 3) * N + n) * KNEI + kk;
  out[base] = ox; out[base + dstr] = oy; out[base + 2 * dstr] = oz;
}

// direct input (B,C,3,N,KNEI), output mean over KNEI -> (B,O,3,N)
template <int O, int C>
__global__ void vn_apply_direct_mean(const float* __restrict__ in,
                                     const float* __restrict__ wf, const float* __restrict__ wd,
                                     const float* __restrict__ stats, float* __restrict__ xout,
                                     int N, int BN) {
  __shared__ float swf[C], swd[C];
  __shared__ float smu, sistd;
  int o = blockIdx.y;
  for (int t = threadIdx.x; t < C; t += blockDim.x) { swf[t] = wf[o * C + t]; swd[t] = wd[o * C + t]; }
  if (threadIdx.x == 0) { smu = stats[o]; sistd = stats[O + o]; }
  __syncthreads();
  int gid = blockIdx.x * blockDim.x + threadIdx.x;
  if (gid >= BN) return;
  int n = gid % N, b = gid / N;
  const size_t dstr = (size_t)N * KNEI;
  float a0 = 0, a1 = 0, a2 = 0;
  for (int kk = 0; kk < KNEI; ++kk) {
    float px = 0, py = 0, pz = 0, dx = 0, dy = 0, dz = 0;
#pragma unroll
    for (int c = 0; c < C; ++c) {
      size_t base = (((size_t)(b * C + c) * 3) * N + n) * KNEI + kk;
      float f0 = in[base], f1 = in[base + dstr], f2 = in[base + 2 * dstr];
      float wa = swf[c], va = swd[c];
      px += wa * f0; py += wa * f1; pz += wa * f2;
      dx += va * f0; dy += va * f1; dz += va * f2;
    }
    float nrm = sqrtf(px * px + py * py + pz * pz) + EPSV;
    float sc2 = (nrm - smu) * sistd / nrm;
    px *= sc2; py *= sc2; pz *= sc2;
    float ox, oy, oz;
    vn_lrelu3(px, py, pz, dx, dy, dz, ox, oy, oz);
    a0 += ox; a1 += oy; a2 += oz;
  }
  float inv = 1.f / KNEI;
  size_t ob = ((size_t)(b * O + o) * 3) * N + n;
  xout[ob] = a0 * inv; xout[ob + N] = a1 * inv; xout[ob + 2 * N] = a2 * inv;
}

// graph-feature input, output mean over KNEI -> (B,O,3,N)
template <int O, int C>
__global__ void vn_apply_gf_mean(const float* __restrict__ src, const int* __restrict__ idx,
                                 const float* __restrict__ wf, const float* __restrict__ wd,
                                 const float* __restrict__ stats, float* __restrict__ xout,
                                 int N, int BN) {
  const int C2 = 2 * C;
  __shared__ float swf[C2], swd[C2];
  __shared__ float smu, sistd;
  int o = blockIdx.y;
  for (int t = threadIdx.x; t < C2; t += blockDim.x) { swf[t] = wf[o * C2 + t]; swd[t] = wd[o * C2 + t]; }
  if (threadIdx.x == 0) { smu = stats[o]; sistd = stats[O + o]; }
  __syncthreads();
  int gid = blockIdx.x * blockDim.x + threadIdx.x;
  if (gid >= BN) return;
  int n = gid % N, b = gid / N;
  float a0 = 0, a1 = 0, a2 = 0;
  for (int kk = 0; kk < KNEI; ++kk) {
    int nbr = idx[(size_t)gid * KNEI + kk];
    float px = 0, py = 0, pz = 0, dx = 0, dy = 0, dz = 0;
#pragma unroll
    for (int c = 0; c < C; ++c) {
      const float* sc = src + ((size_t)(b * C + c) * 3) * N;
      float c0 = sc[n],   c1 = sc[N + n],   c2 = sc[2 * N + n];
      float f0 = sc[nbr] - c0, f1 = sc[N + nbr] - c1, f2 = sc[2 * N + nbr] - c2;
      float wa = swf[c], wb = swf[C + c], va = swd[c], vb = swd[C + c];
      px += wa * f0 + wb * c0; py += wa * f1 + wb * c1; pz += wa * f2 + wb * c2;
      dx += va * f0 + vb * c0; dy += va * f1 + vb * c1; dz += va * f2 + vb * c2;
    }
    float nrm = sqrtf(px * px + py * py + pz * pz) + EPSV;
    float sc2 = (nrm - smu) * sistd / nrm;
    px *= sc2; py *= sc2; pz *= sc2;
    float ox, oy, oz;
    vn_lrelu3(px, py, pz, dx, dy, dz, ox, oy, oz);
    a0 += ox; a1 += oy; a2 += oz;
  }
  float inv = 1.f / KNEI;
  size_t ob = ((size_t)(b * O + o) * 3) * N + n;
  xout[ob] = a0 * inv; xout[ob + N] = a1 * inv; xout[ob + 2 * N] = a2 * inv;
}

// ---------------- layer 6: p6 = w6f(1024x63) @ x123 via WMMA f32 ----------------
__global__ void gemm6_wmma(const float* __restrict__ x1, const float* __restrict__ x2,
                           const float* __restrict__ x3, const float* __restrict__ w6f,
                           float* __restrict__ p6, int N) {
  __shared__ float ldsA[16][64];  // [o][c]
  __shared__ float ldsB[16][64];  // [pos][c]
  int lane = threadIdx.x;
  int pos0 = blockIdx.x << 4;     // pos = (b*3+d)*N + n
  int o0   = blockIdx.y << 4;
  int bd = pos0 / N;
  int n0 = pos0 % N;
  int b = bd / 3, d = bd % 3;

  for (int t = lane; t < 16 * 63; t += 32) {
    int o = t / 63, c = t % 63;
    ldsA[o][c] = w6f[(size_t)(o0 + o) * 63 + c];
  }
  if (lane < 16) ldsA[lane][63] = 0.f;
  for (int t = lane; t < 16 * 64; t += 32) {
    int c = t >> 4, p = t & 15;
    float v = 0.f;
    if (c < 63) {
      int cc = c % 21;
      const float* src = (c < 21) ? x1 : ((c < 42) ? x2 : x3);
      v = src[(((size_t)(b * 21 + cc) * 3) + d) * N + n0 + p];
    }
    ldsB[p][c] = v;
  }
  __syncthreads();

  v8f acc = {};
  int row  = lane & 15;
  int koff = (lane >> 4) << 1;
#pragma unroll
  for (int kk = 0; kk < 16; ++kk) {
    v2f a, bb;
    a.x  = ldsA[row][4 * kk + koff];  a.y  = ldsA[row][4 * kk + koff + 1];
    bb.x = ldsB[row][4 * kk + koff];  bb.y = ldsB[row][4 * kk + koff + 1];
    acc = wmma_f32_k4(a, bb, acc);
  }
#pragma unroll
  for (int r = 0; r < 8; ++r) {
    int o = o0 + r + ((lane >= 16) ? 8 : 0);
    p6[(((size_t)(b * FF + o) * 3) + d) * N + n0 + (lane & 15)] = acc[r];
  }
}

__global__ void d6_kernel(const float* __restrict__ x1, const float* __restrict__ x2,
                          const float* __restrict__ x3, const float* __restrict__ w6d,
                          float* __restrict__ d6, int N, int total) {
  int gid = blockIdx.x * blockDim.x + threadIdx.x;
  if (gid >= total) return;
  int n = gid % N;
  int d = (gid / N) % 3;
  int b = gid / (3 * N);
  float s = 0.f;
  for (int c = 0; c < 63; ++c) {
    int cc = c % 21;
    const float* src = (c < 21) ? x1 : ((c < 42) ? x2 : x3);
    s += w6d[c] * src[(((size_t)(b * 21 + cc) * 3) + d) * N + n];
  }
  d6[gid] = s;
}

__global__ void stats6_kernel(const float* __restrict__ p6, float* __restrict__ stats,
                              int N, int B, float cntInv) {
  int o = blockIdx.x;
  __shared__ float r1[256], r2[256];
  float s1 = 0.f, s2 = 0.f;
  for (int i = threadIdx.x; i < B * N; i += 256) {
    int b = i / N, n = i % N;
    size_t base = ((size_t)(b * FF + o) * 3) * N;
    float px = p6[base + n], py = p6[base + N + n], pz = p6[base + 2 * N + n];
    float nrm = sqrtf(px * px + py * py + pz * pz) + EPSV;
    s1 += nrm; s2 += nrm * nrm;
  }
  r1[threadIdx.x] = s1; r2[threadIdx.x] = s2; __syncthreads();
  for (int s = 128; s > 0; s >>= 1) {
    if (threadIdx.x < s) { r1[threadIdx.x] += r1[threadIdx.x + s]; r2[threadIdx.x] += r2[threadIdx.x + s]; }
    __syncthreads();
  }
  if (threadIdx.x == 0) {
    float mu = r1[0] * cntInv;
    float var = r2[0] * cntInv - mu * mu;
    stats[o] = mu;
    stats[FF + o] = rsqrtf(var + BN_EPSV);
  }
}

// bn + lrelu + mean over n -> m (B,F,3)
__global__ void apply6_kernel(const float* __restrict__ p6, const float* __restrict__ d6,
                              const float* __restrict__ stats, float* __restrict__ m, int N) {
  int b = blockIdx.x >> 10;
  int o = blockIdx.x & 1023;
  float mu = stats[o], istd = stats[FF + o];
  size_t base  = ((size_t)(b * FF + o) * 3) * N;
  size_t dbase = ((size_t)b * 3) * N;
  float a0 = 0, a1 = 0, a2 = 0;
  for (int n = threadIdx.x; n < N; n += blockDim.x) {
    float px = p6[base + n], py = p6[base + N + n], pz = p6[base + 2 * N + n];
    float dx = d6[dbase + n], dy = d6[dbase + N + n], dz = d6[dbase + 2 * N + n];
    float nrm = sqrtf(px * px + py * py + pz * pz) + EPSV;
    float sc = (nrm - mu) * istd / nrm;
    px *= sc; py *= sc; pz *= sc;
    float ox, oy, oz;
    vn_lrelu3(px, py, pz, dx, dy, dz, ox, oy, oz);
    a0 += ox; a1 += oy; a2 += oz;
  }
  __shared__ float r[3 * 128];
  r[threadIdx.x] = a0; r[128 + threadIdx.x] = a1; r[256 + threadIdx.x] = a2;
  __syncthreads();
  for (int s = 64; s > 0; s >>= 1) {
    if (threadIdx.x < s) {
      r[threadIdx.x] += r[threadIdx.x + s];
      r[128 + threadIdx.x] += r[128 + threadIdx.x + s];
      r[256 + threadIdx.x] += r[256 + threadIdx.x + s];
    }
    __syncthreads();
  }
  if (threadIdx.x == 0) {
    float inv = 1.f / N;
    size_t mb = (size_t)(b * FF + o) * 3;
    m[mb] = r[0] * inv; m[mb + 1] = r[128] * inv; m[mb + 2] = r[256] * inv;
  }
}

// ---------------- head: dense over channels, per 3-vector component ----------------
// out[b,o,d] = sum_c (W[o,c] (+ W[o,Cin+c] if fold)) * in[b,c,d]
__global__ void dense3_kernel(const float* __restrict__ W, const float* __restrict__ in,
                              float* __restrict__ out, int O, int Cin, int fold, int total) {
  int gid = blockIdx.x * blockDim.x + threadIdx.x;
  if (gid >= total) return;
  int d = gid % 3;
  int o = (gid / 3) % O;
  int b = gid / (3 * O);
  const float* wr = W + (size_t)o * (fold ? 2 * Cin : Cin);
  const float* ib = in + ((size_t)b * Cin) * 3 + d;
  float s = 0.f;
  for (int c = 0; c < Cin; ++c) {
    float w = fold ? (wr[c] + wr[Cin + c]) : wr[c];
    s += w * ib[(size_t)c * 3];
  }
  out[gid] = s;
}

// Gram-Schmidt of (j, j_) -> R (B,F,3,3), plus v_loc = R^T v
__global__ void gram_vloc_kernel(const float* __restrict__ j, const float* __restrict__ j2,
                                 const float* __restrict__ v, float* __restrict__ R,
                                 float* __restrict__ vloc, int total) {
  int gid = blockIdx.x * blockDim.x + threadIdx.x;
  if (gid >= total) return;
  size_t b3 = (size_t)gid * 3;
  float a1x = j[b3], a1y = j[b3 + 1], a1z = j[b3 + 2];
  float a2x = j2[b3], a2y = j2[b3 + 1], a2z = j2[b3 + 2];
  float n1 = sqrtf(a1x * a1x + a1y * a1y + a1z * a1z) + 1e-8f;
  float b1x = a1x / n1, b1y = a1y / n1, b1z = a1z / n1;
  float dp = b1x * a2x + b1y * a2y + b1z * a2z;
  float px = a2x - dp * b1x, py = a2y - dp * b1y, pz = a2z - dp * b1z;
  float n2 = sqrtf(px * px + py * py + pz * pz) + 1e-8f;
  float b2x = px / n2, b2y = py / n2, b2z = pz / n2;
  float b3x = b1y * b2z - b1z * b2y;
  float b3y = b1z * b2x - b1x * b2z;
  float b3z = b1x * b2y - b1y * b2x;
  size_t rb = (size_t)gid * 9;  // R[d][k], columns = b1,b2,b3
  R[rb + 0] = b1x; R[rb + 1] = b2x; R[rb + 2] = b3x;
  R[rb + 3] = b1y; R[rb + 4] = b2y; R[rb + 5] = b3y;
  R[rb + 6] = b1z; R[rb + 7] = b2z; R[rb + 8] = b3z;
  float vx = v[b3], vy = v[b3 + 1], vz = v[b3 + 2];
  vloc[b3 + 0] = b1x * vx + b1y * vy + b1z * vz;
  vloc[b3 + 1] = b2x * vx + b2y * vy + b2z * vz;
  vloc[b3 + 2] = b3x * vx + b3y * vy + b3z * vz;
}

// comb[b,o,d] = v[b,o,d] + sum_k R[b,o,d,k] * h[b,o,k]
__global__ void rotout_kernel(const float* __restrict__ R, const float* __restrict__ h,
                              const float* __restrict__ v, float* __restrict__ out, int total) {
  int gid = blockIdx.x * blockDim.x + threadIdx.x;
  if (gid >= total) return;
  size_t b3 = (size_t)gid * 3, rb = (size_t)gid * 9;
  float h0 = h[b3], h1 = h[b3 + 1], h2 = h[b3 + 2];
#pragma unroll
  for (int d = 0; d < 3; ++d) {
    out[b3 + d] = v[b3 + d] + R[rb + d * 3 + 0] * h0 + R[rb + d * 3 + 1] * h1 + R[rb + d * 3 + 2] * h2;
  }
}

// ---------------- orchestration ----------------
extern "C" void kernel_launch(void* const* d_in, const int* in_sizes, int n_in,
                              void* d_out, int out_size, void* d_ws, size_t ws_size,
                              hipStream_t stream) {
  (void)in_sizes; (void)n_in; (void)out_size; (void)ws_size;
  const float* x    = (const float*)d_in[0];
  const float* w1f  = (const float*)d_in[1];
  const float* w1d  = (const float*)d_in[2];
  const float* w2f  = (const float*)d_in[3];
  const float* w2d  = (const float*)d_in[4];
  const float* w3f  = (const float*)d_in[5];
  const float* w3d  = (const float*)d_in[6];
  const float* w4f  = (const float*)d_in[7];
  const float* w4d  = (const float*)d_in[8];
  const float* w5f  = (const float*)d_in[9];
  const float* w5d  = (const float*)d_in[10];
  const float* w6f  = (const float*)d_in[11];
  const float* w6d  = (const float*)d_in[12];
  const float* wl1  = (const float*)d_in[13];
  const float* wl2  = (const float*)d_in[14];
  const float* wl2b = (const float*)d_in[15];
  const float* wl3  = (const float*)d_in[16];
  const float* wl4  = (const float*)d_in[17];
  const float* wl4b = (const float*)d_in[18];
  const float* wl5  = (const float*)d_in[19];
  const float* wl6  = (const float*)d_in[20];
  const float* wl6b = (const float*)d_in[21];
  const float* wc1  = (const float*)d_in[22];
  const float* wc2  = (const float*)d_in[23];
  const float* wc3  = (const float*)d_in[24];

  const int B = BB, N = NN;
  const int P  = B * N * KNEI;     // 163840
  const int BN = B * N;            // 8192
  const int NCH = 64;              // reduction chunks
  const int CHS = P / NCH;         // 2560

  // workspace carving (256B aligned)
  char* wp = (char*)d_ws;
  auto carve = [&](size_t bytes) -> void* {
    void* r = (void*)wp;
    wp += (bytes + 255) & ~(size_t)255;
    return r;
  };
  const size_t szH  = (size_t)B * 21 * 3 * N * KNEI;   // floats
  const size_t szP6 = (size_t)B * FF * 3 * N;          // floats
  int*   idxB  = (int*)  carve((size_t)P * 4);
  float* xx    = (float*)carve((size_t)BN * 4);
  float* big   = (float*)carve((2 * szH > szP6 ? 2 * szH : szP6) * 4);
  float* hA = big;            // layer1 out  (aliased with p6; lifetimes disjoint)
  float* hB = big + szH;      // layer3 out
  float* p6 = big;
  float* x1    = (float*)carve((size_t)B * 21 * 3 * N * 4);
  float* x2    = (float*)carve((size_t)B * 21 * 3 * N * 4);
  float* x3    = (float*)carve((size_t)B * 21 * 3 * N * 4);
  float* part  = (float*)carve((size_t)21 * NCH * 2 * 4);
  float* stats = (float*)carve((size_t)2 * FF * 4);
  float* d6b   = (float*)carve((size_t)B * 3 * N * 4);
  float* mbuf  = (float*)carve((size_t)B * FF * 3 * 4);
  float* vb    = (float*)carve((size_t)B * FF * 3 * 4);
  float* jb    = (float*)carve((size_t)B * FF * 3 * 4);
  float* j2b   = (float*)carve((size_t)B * FF * 3 * 4);
  float* vloc  = (float*)carve((size_t)B * FF * 3 * 4);
  float* hmix  = (float*)carve((size_t)B * FF * 3 * 4);
  float* Rb    = (float*)carve((size_t)B * FF * 9 * 4);
  float* comb  = (float*)carve((size_t)B * FF * 3 * 4);

  const int knnBlocks = B * (N / 16);
  const float invP  = 1.f / (float)P;
  const float invBN = 1.f / (float)BN;

  // ---- stage 1 (C=1) ----
  sqnorm_kernel<<<(BN + 255) / 256, 256, 0, stream>>>(x, xx, 3, N, BN);
  knn_topk_kernel<<<knnBlocks, 32, 0, stream>>>(x, xx, idxB, 3, N);
  vn_stats_gf<21, 1><<<dim3(NCH, 21), 256, 0, stream>>>(x, idxB, w1f, part, N, P, CHS);
  finalize_stats<<<1, 32, 0, stream>>>(part, stats, 21, NCH, invP);
  vn_apply_gf_full<21, 1><<<dim3((P + 255) / 256, 21), 256, 0, stream>>>(x, idxB, w1f, w1d, stats, hA, N, P);
  // ---- stage 2 (direct 21->21), mean -> x1 ----
  vn_stats_direct<21, 21><<<dim3(NCH, 21), 256, 0, stream>>>(hA, w2f, part, N, P, CHS);
  finalize_stats<<<1, 32, 0, stream>>>(part, stats, 21, NCH, invP);
  vn_apply_direct_mean<21, 21><<<dim3((BN + 255) / 256, 21), 256, 0, stream>>>(hA, w2f, w2d, stats, x1, N, BN);
  // ---- knn on x1 (D=63) ----
  sqnorm_kernel<<<(BN + 255) / 256, 256, 0, stream>>>(x1, xx, 63, N, BN);
  knn_topk_kernel<<<knnBlocks, 32, 0, stream>>>(x1, xx, idxB, 63, N);
  // ---- stage 3 (gf 42->21, full -> hB) ----
  vn_stats_gf<21, 21><<<dim3(NCH, 21), 256, 0, stream>>>(x1, idxB, w3f, part, N, P, CHS);
  finalize_stats<<<1, 32, 0, stream>>>(part, stats, 21, NCH, invP);
  vn_apply_gf_full<21, 21><<<dim3((P + 255) / 256, 21), 256, 0, stream>>>(x1, idxB, w3f, w3d, stats, hB, N, P);
  // ---- stage 4 (direct), mean -> x2 ----
  vn_stats_direct<21, 21><<<dim3(NCH, 21), 256, 0, stream>>>(hB, w4f, part, N, P, CHS);
  finalize_stats<<<1, 32, 0, stream>>>(part, stats, 21, NCH, invP);
  vn_apply_direct_mean<21, 21><<<dim3((BN + 255) / 256, 21), 256, 0, stream>>>(hB, w4f, w4d, stats, x2, N, BN);
  // ---- knn on x2 ----
  sqnorm_kernel<<<(BN + 255) / 256, 256, 0, stream>>>(x2, xx, 63, N, BN);
  knn_topk_kernel<<<knnBlocks, 32, 0, stream>>>(x2, xx, idxB, 63, N);
  // ---- stage 5 (gf 42->21), mean -> x3 ----
  vn_stats_gf<21, 21><<<dim3(NCH, 21), 256, 0, stream>>>(x2, idxB, w5f, part, N, P, CHS);
  finalize_stats<<<1, 32, 0, stream>>>(part, stats, 21, NCH, invP);
  vn_apply_gf_mean<21, 21><<<dim3((BN + 255) / 256, 21), 256, 0, stream>>>(x2, idxB, w5f, w5d, stats, x3, N, BN);
  // ---- stage 6 (1024x63 WMMA GEMM + BN + lrelu + mean over N) ----
  gemm6_wmma<<<dim3(B * 3 * N / 16, FF / 16), 32, 0, stream>>>(x1, x2, x3, w6f, p6, N);
  d6_kernel<<<(B * 3 * N + 255) / 256, 256, 0, stream>>>(x1, x2, x3, w6d, d6b, N, B * 3 * N);
  stats6_kernel<<<FF, 256, 0, stream>>>(p6, stats, N, B, invBN);
  apply6_kernel<<<B * FF, 128, 0, stream>>>(p6, d6b, stats, mbuf, N);
  // ---- head: 3 residual geometric blocks ----
  const int T3 = B * FF * 3;
  const int TB = B * FF;
  // block 1 (fold: hfin = [m; m])
  dense3_kernel<<<(T3 + 255) / 256, 256, 0, stream>>>(wl1,  mbuf, vb,  FF, FF, 1, T3);
  dense3_kernel<<<(T3 + 255) / 256, 256, 0, stream>>>(wl2,  mbuf, jb,  FF, FF, 1, T3);
  dense3_kernel<<<(T3 + 255) / 256, 256, 0, stream>>>(wl2b, mbuf, j2b, FF, FF, 1, T3);
  gram_vloc_kernel<<<(TB + 255) / 256, 256, 0, stream>>>(jb, j2b, vb, Rb, vloc, TB);
  dense3_kernel<<<(T3 + 255) / 256, 256, 0, stream>>>(wc1, vloc, hmix, FF, FF, 0, T3);
  rotout_kernel<<<(TB + 255) / 256, 256, 0, stream>>>(Rb, hmix, vb, comb, TB);
  // block 2
  dense3_kernel<<<(T3 + 255) / 256, 256, 0, stream>>>(wl3,  comb, vb,  FF, FF, 0, T3);
  dense3_kernel<<<(T3 + 255) / 256, 256, 0, stream>>>(wl4,  comb, jb,  FF, FF, 0, T3);
  dense3_kernel<<<(T3 + 255) / 256, 256, 0, stream>>>(wl4b, comb, j2b, FF, FF, 0, T3);
  gram_vloc_kernel<<<(TB + 255) / 256, 256, 0, stream>>>(jb, j2b, vb, Rb, vloc, TB);
  dense3_kernel<<<(T3 + 255) / 256, 256, 0, stream>>>(wc2, vloc, hmix, FF, FF, 0, T3);
  rotout_kernel<<<(TB + 255) / 256, 256, 0, stream>>>(Rb, hmix, vb, comb, TB);
  // block 3 -> d_out
  dense3_kernel<<<(T3 + 255) / 256, 256, 0, stream>>>(wl5,  comb, vb,  FF, FF, 0, T3);
  dense3_kernel<<<(T3 + 255) / 256, 256, 0, stream>>>(wl6,  comb, jb,  FF, FF, 0, T3);
  dense3_kernel<<<(T3 + 255) / 256, 256, 0, stream>>>(wl6b, comb, j2b, FF, FF, 0, T3);
  gram_vloc_kernel<<<(TB + 255) / 256, 256, 0, stream>>>(jb, j2b, vb, Rb, vloc, TB);
  dense3_kernel<<<(T3 + 255) / 256, 256, 0, stream>>>(wc3, vloc, hmix, FF, FF, 0, T3);
  rotout_kernel<<<(TB + 255) / 256, 256, 0, stream>>>(Rb, hmix, vb, (float*)d_out, TB);
}